// NeuralMeshSimplification_36361193128540
// MI455X (gfx1250) — compile-verified
//
#include <hip/hip_runtime.h>
#include <math.h>

// ---------------------------------------------------------------------------
// NeuralMeshSimplification for MI455X (gfx1250, wave32).
// Dominant compute: fd2 = pairwise dist of adj rows -> adj @ adjT (symmetric)
//   = 2*4000^3 = 128 GFLOP fp32. adj padded to 4096x4096 f32 (67MB) fits in
//   the 192MB L2 -> compute bound; done with V_WMMA_F32_16X16X4_F32, fused
//   with the per-row top-16 selection so fd2 is never materialized.
//   K-tiles staged into LDS with double-buffered GLOBAL_LOAD_ASYNC_TO_LDS_B128
//   (ASYNCcnt) so the DMA of tile k+1 overlaps the 16 WMMAs of tile k and the
//   data never round-trips through VGPRs.
// Second GEMM: triangle MLP 480000 x 67 x 128 (8.2 GFLOP), also f32 WMMA.
// All other stages are <100MB of traffic (us at 23.3 TB/s).
// ---------------------------------------------------------------------------

typedef __attribute__((ext_vector_type(2))) float v2f;
typedef __attribute__((ext_vector_type(8))) float v8f;
typedef __attribute__((ext_vector_type(4))) int v4i;

#define NPTS  8000
#define DIM   64
#define HID   128
#define NS    4000
#define EK    15
#define KNN   16
#define NPAIR 120           // K*(K-1)/2 with K=16
#define NT    (NS * NPAIR)  // 480000 triangles
#define ADJS  4096          // padded adj stride (zeros beyond 4000)

__device__ __forceinline__ float sigmoidf(float v) { return 1.0f / (1.0f + expf(-v)); }

// ---- async global->LDS copy (CDNA5): guarded so we always compile ----------
#if defined(__HIP_DEVICE_COMPILE__) && __has_builtin(__builtin_amdgcn_global_load_async_to_lds_b128)
#define USE_ASYNC_LDS 1
__device__ __forceinline__ void async_b128(const float* g, float* l) {
  // builtin parameter type is v4i* (per hipcc diagnostic): (global_src, lds_dst, offset, cpol)
  __builtin_amdgcn_global_load_async_to_lds_b128((v4i*)g, (v4i*)l, 0, 0);
}
#if __has_builtin(__builtin_amdgcn_s_wait_asynccnt)
#define WAIT_ASYNC(n) __builtin_amdgcn_s_wait_asynccnt(n)
#else
#define WAIT_ASYNC(n) asm volatile("s_wait_asynccnt %0" ::"i"(n) : "memory")
#endif
#else
#define USE_ASYNC_LDS 0
#define WAIT_ASYNC(n) ((void)0)
#endif

// ---------------- zero fill -------------------------------------------------
__global__ void k_zero(float* __restrict__ p, size_t n) {
  size_t stride = (size_t)gridDim.x * blockDim.x;
  for (size_t s = (size_t)blockIdx.x * blockDim.x + threadIdx.x; s < n; s += stride)
    p[s] = 0.0f;
}

// ---------------- stage 1: probs = sigmoid(relu(x@W1+b1)@w2) ---------------
__global__ __launch_bounds__(HID) void k_probs(const float* __restrict__ x,
                                               const float* __restrict__ W1,
                                               const float* __restrict__ b1,
                                               const float* __restrict__ w2,
                                               float* __restrict__ probs) {
  __shared__ float xs[DIM];
  __shared__ float red[HID];
  int row = blockIdx.x, t = threadIdx.x;
  if (t < DIM) xs[t] = x[(size_t)row * DIM + t];
  __syncthreads();
  float acc = b1[t];
#pragma unroll 8
  for (int d = 0; d < DIM; ++d) acc = fmaf(xs[d], W1[d * HID + t], acc);
  red[t] = fmaxf(acc, 0.0f) * w2[t];
  __syncthreads();
  for (int s = HID / 2; s > 0; s >>= 1) {
    if (t < s) red[t] += red[t + s];
    __syncthreads();
  }
  if (t == 0) probs[row] = sigmoidf(red[0]);
}

// ---------------- top-4000 selection with exact top_k tie semantics --------
__global__ void k_rank(const float* __restrict__ probs, int* __restrict__ selidx) {
  int i = blockIdx.x * blockDim.x + threadIdx.x;
  if (i >= NPTS) return;
  float pi = probs[i];
  int rank = 0;
  for (int j = 0; j < NPTS; ++j) {
    float pj = probs[j];
    rank += (pj > pi) || (pj == pi && j < i);
  }
  if (rank < NS) selidx[rank] = i;  // sorted order: descending prob, ties by index
}

// ---------------- gather sx, sp --------------------------------------------
__global__ void k_gather(const float* __restrict__ x, const float* __restrict__ pos,
                         const int* __restrict__ selidx,
                         float* __restrict__ sx, float* __restrict__ sp) {
  int tid = blockIdx.x * blockDim.x + threadIdx.x;
  if (tid < NS * DIM) {
    int r = tid >> 6, d = tid & 63;
    sx[tid] = x[(size_t)selidx[r] * DIM + d];
  }
  if (tid < NS * 3) {
    int r = tid / 3, d = tid % 3;
    sp[tid] = pos[(size_t)selidx[r] * 3 + d];
  }
}

// ---------------- spatial 15-NN (jnp formula, stable ties) -----------------
__global__ __launch_bounds__(128) void k_knn_sp(const float* __restrict__ sp,
                                                int* __restrict__ nbr) {
  __shared__ float spS[NS * 3];  // 48 KB, whole point cloud in LDS
  int t = threadIdx.x;
  for (int s = t; s < NS * 3; s += 128) spS[s] = sp[s];
  __syncthreads();
  int r = blockIdx.x * 128 + t;
  if (r >= NS) return;
  float px = spS[r * 3], py = spS[r * 3 + 1], pz = spS[r * 3 + 2];
  float sr = px * px + py * py + pz * pz;
  float bv[EK];
  int bj[EK];
#pragma unroll
  for (int q = 0; q < EK; ++q) { bv[q] = 3.0e38f; bj[q] = 0; }
  for (int j = 0; j < NS; ++j) {
    if (j == r) continue;
    float qx = spS[j * 3], qy = spS[j * 3 + 1], qz = spS[j * 3 + 2];
    float d2 = sr + qx * qx + qy * qy + qz * qz - 2.0f * (px * qx + py * qy + pz * qz);
    if (d2 < bv[EK - 1]) {  // strict <: earlier index wins ties (top_k order)
      int p = EK - 1;
      while (p > 0 && bv[p - 1] > d2) { bv[p] = bv[p - 1]; bj[p] = bj[p - 1]; --p; }
      bv[p] = d2; bj[p] = j;
    }
  }
  for (int q = 0; q < EK; ++q) nbr[r * EK + q] = bj[q];
}

// ---------------- edge probabilities (one wave32 per edge) -----------------
__global__ __launch_bounds__(256) void k_edges(const float* __restrict__ sx,
                                               const float* __restrict__ w_e,
                                               const int* __restrict__ nbr,
                                               float* __restrict__ adj) {
  int tid = blockIdx.x * 256 + threadIdx.x;
  int wv = tid >> 5, lane = tid & 31;
  if (wv >= NS * EK) return;
  int i = wv / EK, k = wv % EK;
  int j = nbr[i * EK + k];
  float s = sx[(size_t)i * DIM + lane] * sx[(size_t)j * DIM + lane] * w_e[lane] +
            sx[(size_t)i * DIM + lane + 32] * sx[(size_t)j * DIM + lane + 32] * w_e[lane + 32];
#pragma unroll
  for (int off = 16; off > 0; off >>= 1) s += __shfl_xor(s, off, 32);
  if (lane == 0) {
    float e = sigmoidf(s);  // symmetric in (i,j): benign identical-value race
    adj[(size_t)i * ADJS + j] = e;
    adj[(size_t)j * ADJS + i] = e;
  }
}

// ---------------- row norms of adj -----------------------------------------
__global__ __launch_bounds__(256) void k_norms(const float* __restrict__ adj,
                                               float* __restrict__ norms) {
  __shared__ float red[256];
  int r = blockIdx.x, t = threadIdx.x;
  float s = 0.0f;
  for (int c = t; c < ADJS; c += 256) {
    float v = adj[(size_t)r * ADJS + c];
    s = fmaf(v, v, s);
  }
  red[t] = s;
  __syncthreads();
  for (int w = 128; w > 0; w >>= 1) {
    if (t < w) red[t] += red[t + w];
    __syncthreads();
  }
  if (t == 0) norms[r] = red[0];
}

// ---------------- fd2 GEMM (adj @ adjT via symmetry) fused with top-16 -----
// Workgroup = 4 waves = 64-row block; wave w owns rows [w*16,w*16+16) x 64 cols
// (4 accumulators). j-blocks of 64 streamed; K tiled by 16 through LDS with
// double-buffered async global->LDS DMA.
__global__ __launch_bounds__(128) void k_fd2knn(const float* __restrict__ adj,
                                                const float* __restrict__ norms,
                                                int* __restrict__ knn) {
  __shared__ __align__(16) float As[2][64 * 16];
  __shared__ __align__(16) float Bs[2][16 * 64];
  __shared__ float Ct[64 * 64];
  __shared__ float normJ[64];
  int t = threadIdx.x;
  int wave = t >> 5, lane = t & 31;
  int mrow = lane & 15;
  int khalf = (lane >> 4) * 2;  // A/B 2-VGPR layout: K=0/1 lanes 0-15, K=2/3 lanes 16-31
  int i0 = blockIdx.x * 64;
  int gi = i0 + t;  // selection row for t < 64
  float si = (t < 64 && gi < NS) ? norms[gi] : 0.0f;
  float bestv[KNN];
  int bestj[KNN];
#pragma unroll
  for (int q = 0; q < KNN; ++q) { bestv[q] = 3.0e38f; bestj[q] = 0; }

  for (int jb = 0; jb < 63; ++jb) {
    int j0 = jb * 64;
    v8f zero = {};
    v8f cacc[4];
#pragma unroll
    for (int q = 0; q < 4; ++q) cacc[q] = zero;

    // stage one 16-deep K-slab of A (64x16) and B (16x64) into buffer `buf`
    auto stage = [&](int buf, int kk) {
      int k0 = kk * 16;
#if USE_ASYNC_LDS
#pragma unroll
      for (int q = 0; q < 2; ++q) {
        int s = t + q * 128;  // one B128 = 4 floats; 256 chunks per tile
        async_b128(adj + (size_t)(i0 + (s >> 2)) * ADJS + k0 + (s & 3) * 4,
                   &As[buf][s * 4]);
        async_b128(adj + (size_t)(k0 + (s >> 4)) * ADJS + j0 + (s & 15) * 4,
                   &Bs[buf][s * 4]);
      }
#else
      for (int s = t; s < 64 * 16; s += 128)
        As[buf][s] = adj[(size_t)(i0 + (s >> 4)) * ADJS + (k0 + (s & 15))];
      for (int s = t; s < 16 * 64; s += 128)
        Bs[buf][s] = adj[(size_t)(k0 + (s >> 6)) * ADJS + (j0 + (s & 63))];
#endif
    };

    stage(0, 0);
    for (int kk = 0; kk < 250; ++kk) {
      int cur = kk & 1;
      if (kk + 1 < 250) {  // prefetch next slab into the other buffer
        stage(cur ^ 1, kk + 1);
        WAIT_ASYNC(4);     // 4 newest in flight; everything older (cur) done
      } else {
        WAIT_ASYNC(0);
      }
      __syncthreads();
#pragma unroll
      for (int ks = 0; ks < 4; ++ks) {
        int kb = ks * 4 + khalf;
        v2f a;
        a.x = As[cur][(wave * 16 + mrow) * 16 + kb];
        a.y = As[cur][(wave * 16 + mrow) * 16 + kb + 1];
#pragma unroll
        for (int ct = 0; ct < 4; ++ct) {
          v2f b;
          b.x = Bs[cur][kb * 64 + ct * 16 + mrow];
          b.y = Bs[cur][(kb + 1) * 64 + ct * 16 + mrow];
          cacc[ct] = __builtin_amdgcn_wmma_f32_16x16x4_f32(
              false, a, false, b, (short)0, cacc[ct], false, false);
        }
      }
      __syncthreads();
    }

    // dump dot-products tile to LDS (C/D layout: M = r + 8*(lane/16), N = lane%16)
#pragma unroll
    for (int ct = 0; ct < 4; ++ct) {
#pragma unroll
      for (int r = 0; r < 8; ++r) {
        int rowl = wave * 16 + r + 8 * (lane >> 4);
        Ct[rowl * 64 + ct * 16 + mrow] = cacc[ct][r];
      }
    }
    if (t < 64) normJ[t] = (j0 + t < NS) ? norms[j0 + t] : 0.0f;
    __syncthreads();

    if (t < 64 && gi < NS) {
      for (int n = 0; n < 64; ++n) {
        int gj = j0 + n;
        if (gj >= NS || gj == gi) continue;  // diag masked to +inf in reference
        float f = si + normJ[n] - 2.0f * Ct[t * 64 + n];
        if (f < bestv[KNN - 1]) {
          int p = KNN - 1;
          while (p > 0 && bestv[p - 1] > f) {
            bestv[p] = bestv[p - 1]; bestj[p] = bestj[p - 1]; --p;
          }
          bestv[p] = f; bestj[p] = gj;
        }
      }
    }
    __syncthreads();
  }
  if (t < 64 && gi < NS)
    for (int q = 0; q < KNN; ++q) knn[gi * KNN + q] = bestj[q];
}

// ---------------- triangle candidates + tprob ------------------------------
__global__ void k_tri(const float* __restrict__ adj, const int* __restrict__ knn,
                      float* __restrict__ tprob, float* __restrict__ validf,
                      int* __restrict__ n1F, int* __restrict__ n2F) {
  int g = blockIdx.x * blockDim.x + threadIdx.x;
  if (g >= NT) return;
  int i = g / NPAIR, tt = g % NPAIR;
  int rem = tt, jj = 0, cnt = KNN - 1;  // triu_indices(16,1) row-major order
  while (rem >= cnt) { rem -= cnt; ++jj; --cnt; }
  int ll = jj + 1 + rem;
  int n1 = knn[i * KNN + jj], n2 = knn[i * KNN + ll];
  float a12 = adj[(size_t)n1 * ADJS + n2];
  float prod = adj[(size_t)i * ADJS + n1] * adj[(size_t)i * ADJS + n2] * a12;
  bool valid = a12 > 0.0f;
  tprob[g] = valid ? cbrtf(fmaxf(prod, 1e-9f)) : 0.0f;
  validf[g] = valid ? 1.0f : 0.0f;
  n1F[g] = n1; n2F[g] = n2;
}

// ---------------- triangle MLP: 16-tri blocks, WMMA f32, K=67 padded to 68 -
__global__ __launch_bounds__(128) void k_trimlp(const float* __restrict__ sx,
                                                const float* __restrict__ sp,
                                                const int* __restrict__ n1F,
                                                const int* __restrict__ n2F,
                                                const float* __restrict__ validf,
                                                const float* __restrict__ Wf1,
                                                const float* __restrict__ bf1,
                                                const float* __restrict__ wf2,
                                                float* __restrict__ fprob) {
  __shared__ float Af[16 * 68];
  __shared__ float Ws[68 * HID];  // 34 KB: Wf1 (+zero pad row 67)
  __shared__ float red[16 * 4];
  int t = threadIdx.x, wave = t >> 5, lane = t & 31;
  int mrow = lane & 15;
  int khalf = (lane >> 4) * 2;
  int g0 = blockIdx.x * 16;

  for (int s = t; s < 68 * HID; s += 128) {
    int r = s >> 7, c = s & 127;
    Ws[s] = (r < 67) ? Wf1[r * HID + c] : 0.0f;
  }
  for (int s = t; s < 16 * 68; s += 128) {
    int m = s / 68, d = s % 68;
    int g = g0 + m;
    int i = g / NPAIR, a = n1F[g], b = n2F[g];
    float v;
    if (d < DIM)
      v = (sx[(size_t)i * DIM + d] + sx[(size_t)a * DIM + d] + sx[(size_t)b * DIM + d]) * (1.0f / 3.0f);
    else if (d < 67) {
      int dd = d - DIM;
      v = (sp[i * 3 + dd] + sp[a * 3 + dd] + sp[b * 3 + dd]) * (1.0f / 3.0f);
    } else
      v = 0.0f;
    Af[s] = v;
  }
  __syncthreads();

  int colbase = wave * 32;  // each wave owns 32 of the 128 hidden cols
  v8f c0 = {}, c1 = {};
#pragma unroll
  for (int ks = 0; ks < 17; ++ks) {
    int kb = ks * 4 + khalf;
    v2f a;
    a.x = Af[mrow * 68 + kb];
    a.y = Af[mrow * 68 + kb + 1];
    v2f b;
    b.x = Ws[kb * HID + colbase + mrow];
    b.y = Ws[(kb + 1) * HID + colbase + mrow];
    c0 = __builtin_amdgcn_wmma_f32_16x16x4_f32(false, a, false, b, (short)0, c0, false, false);
    b.x = Ws[kb * HID + colbase + 16 + mrow];
    b.y = Ws[(kb + 1) * HID + colbase + 16 + mrow];
    c1 = __builtin_amdgcn_wmma_f32_16x16x4_f32(false, a, false, b, (short)0, c1, false, false);
  }

  // epilogue: relu(h + bf1) * wf2, reduce over 128 cols
  float s[8];
  int col0 = colbase + mrow, col1 = colbase + 16 + mrow;
  float b0 = bf1[col0], w0 = wf2[col0], b1v = bf1[col1], w1 = wf2[col1];
#pragma unroll
  for (int r = 0; r < 8; ++r)
    s[r] = fmaxf(c0[r] + b0, 0.0f) * w0 + fmaxf(c1[r] + b1v, 0.0f) * w1;
#pragma unroll
  for (int off = 1; off < 16; off <<= 1)
#pragma unroll
    for (int r = 0; r < 8; ++r) s[r] += __shfl_xor(s[r], off, 32);
  if (lane == 0)
#pragma unroll
    for (int r = 0; r < 8; ++r) red[r * 4 + wave] = s[r];        // rows 0..7
  if (lane == 16)
#pragma unroll
    for (int r = 0; r < 8; ++r) red[(r + 8) * 4 + wave] = s[r];  // rows 8..15
  __syncthreads();
  if (t < 16) {
    float tot = red[t * 4] + red[t * 4 + 1] + red[t * 4 + 2] + red[t * 4 + 3];
    int g = g0 + t;
    fprob[g] = sigmoidf(tot) * validf[g];
  }
}

// ---------------------------------------------------------------------------
extern "C" void kernel_launch(void* const* d_in, const int* in_sizes, int n_in,
                              void* d_out, int out_size, void* d_ws, size_t ws_size,
                              hipStream_t stream) {
  const float* x   = (const float*)d_in[0];
  const float* pos = (const float*)d_in[1];
  const float* W1  = (const float*)d_in[2];
  const float* b1  = (const float*)d_in[3];
  const float* w2  = (const float*)d_in[4];
  const float* w_e = (const float*)d_in[5];
  const float* Wf1 = (const float*)d_in[6];
  const float* bf1 = (const float*)d_in[7];
  const float* wf2 = (const float*)d_in[8];
  float* out = (float*)d_out;  // [fprob(480000) | tprob(480000)]

  char* ws = (char*)d_ws;
  size_t off = 0;
  auto carve = [&](size_t bytes) -> char* {
    char* p = ws + off;
    off = (off + bytes + 255) & ~(size_t)255;
    return p;
  };
  float* probs  = (float*)carve((size_t)NPTS * 4);
  int*   selidx = (int*)carve((size_t)NS * 4);
  float* sx     = (float*)carve((size_t)NS * DIM * 4);
  float* sp     = (float*)carve((size_t)NS * 3 * 4);
  int*   nbr    = (int*)carve((size_t)NS * EK * 4);
  float* norms  = (float*)carve((size_t)NS * 4);
  int*   knn    = (int*)carve((size_t)NS * KNN * 4);
  int*   n1F    = (int*)carve((size_t)NT * 4);
  int*   n2F    = (int*)carve((size_t)NT * 4);
  float* validf = (float*)carve((size_t)NT * 4);
  float* adj    = (float*)carve((size_t)ADJS * ADJS * 4);  // 67 MB, fits L2

  k_zero<<<4096, 256, 0, stream>>>(adj, (size_t)ADJS * ADJS);
  k_probs<<<NPTS, HID, 0, stream>>>(x, W1, b1, w2, probs);
  k_rank<<<(NPTS + 255) / 256, 256, 0, stream>>>(probs, selidx);
  k_gather<<<(NS * DIM + 255) / 256, 256, 0, stream>>>(x, pos, selidx, sx, sp);
  k_knn_sp<<<(NS + 127) / 128, 128, 0, stream>>>(sp, nbr);
  k_edges<<<(NS * EK * 32 + 255) / 256, 256, 0, stream>>>(sx, w_e, nbr, adj);
  k_norms<<<NS, 256, 0, stream>>>(adj, norms);
  k_fd2knn<<<(NS + 63) / 64, 128, 0, stream>>>(adj, norms, knn);
  k_tri<<<(NT + 255) / 256, 256, 0, stream>>>(adj, knn, out + NT, validf, n1F, n2F);
  k_trimlp<<<NT / 16, 128, 0, stream>>>(sx, sp, n1F, n2F, validf, Wf1, bf1, wf2, out);
  (void)in_sizes; (void)n_in; (void)out_size; (void)ws_size;
}